// HGCN_pyg_55430847922851
// MI455X (gfx1250) — compile-verified
//
#include <hip/hip_runtime.h>
#include <hip/hip_bf16.h>

#define MIN_NORM 1e-15f
#define EPSF     1e-7f

typedef __attribute__((ext_vector_type(2))) float v2f;
typedef __attribute__((ext_vector_type(8))) float v8f;

__device__ __forceinline__ float wsum(float v) {
#pragma unroll
  for (int o = 16; o > 0; o >>= 1) v += __shfl_xor(v, o, 32);
  return v;
}

// ---------------- zero ----------------
__global__ void zero_kernel(float* p, size_t n) {
  size_t i = (size_t)blockIdx.x * blockDim.x + threadIdx.x;
  if (i < n) p[i] = 0.0f;
}

// ------------- bias tangent: ub = logmap0(proj(expmap0(proj_tan0(b)))) -------------
__global__ void bias_kernel(const float* __restrict__ b, float* __restrict__ ub, int D) {
  if (threadIdx.x != 0) return;
  float ss = 0.f;
  for (int d = 1; d < D; ++d) ss += b[d] * b[d];
  float xn = fmaxf(sqrtf(ss), MIN_NORM);
  float sc = sinhf(xn) / xn;
  float ss2 = 0.f;
  for (int d = 1; d < D; ++d) { float e = sc * b[d]; ss2 += e * e; }
  float e0  = sqrtf(fmaxf(1.0f + ss2, EPSF));
  float yn  = fmaxf(sqrtf(ss2), MIN_NORM);
  float tht = fmaxf(e0, 1.0f + EPSF);
  float ac  = logf(tht + sqrtf(fmaxf(tht * tht - 1.0f, MIN_NORM)));
  ub[0] = 0.f;
  for (int d = 1; d < D; ++d) ub[d] = ac * (sc * b[d]) / yn;
}

// ------------- per-node logmap0 scale for x (D=128), wave per node -------------
__global__ void row_scale_kernel(const float* __restrict__ x, float* __restrict__ s1, int N) {
  int node = blockIdx.x * (blockDim.x >> 5) + (threadIdx.x >> 5);
  if (node >= N) return;
  int lane = threadIdx.x & 31;
  const float* xr = x + (size_t)node * 128;
  float ss = 0.f;
#pragma unroll
  for (int j = 0; j < 4; ++j) {
    int d = 1 + lane + 32 * j;
    if (d < 128) { float v = xr[d]; ss += v * v; }
  }
  ss = wsum(ss);
  if (lane == 0) {
    float yn  = fmaxf(sqrtf(ss), MIN_NORM);
    float tht = fmaxf(xr[0], 1.0f + EPSF);
    float ac  = logf(tht + sqrtf(fmaxf(tht * tht - 1.0f, MIN_NORM)));
    s1[node] = ac / yn;
  }
}

// ------------- GEMM1: z1 = (logmap0(x)) @ W1^T via fp32 WMMA (row-scale folded) -------------
__global__ void gemm1_kernel(const float* __restrict__ x, const float* __restrict__ W1,
                             const float* __restrict__ s1, float* __restrict__ z1, int N) {
  int m0   = blockIdx.x * 16;
  int lane = threadIdx.x;
  int half = lane >> 4;
  int l    = lane & 15;
  int row  = m0 + l;
  int rowc = row < N ? row : N - 1;
  const float* xr = x + (size_t)rowc * 128;
  v8f acc[4] = {};
  // Peeled k == 0: column 0 of logmap0(x) is exactly zero. Only the half==0
  // lane-slot touches kA==0, so mask with a branch-free multiply (xr[0] is
  // valid memory: the time coordinate). Keeps EXEC untouched in the hot loop.
  {
    const float zm = (half == 0) ? 0.0f : 1.0f;
    int kA = 2 * half;
    v2f a;
    a.x = xr[kA] * zm;
    a.y = xr[kA + 1];
#pragma unroll
    for (int nb = 0; nb < 4; ++nb) {
      int col = nb * 16 + l;
      v2f b;
      b.x = W1[col * 128 + kA];
      b.y = W1[col * 128 + kA + 1];
      acc[nb] = __builtin_amdgcn_wmma_f32_16x16x4_f32(false, a, false, b, (short)0,
                                                      acc[nb], false, false);
    }
  }
#pragma unroll 2
  for (int k = 4; k < 128; k += 4) {
    int kA = k + 2 * half;
    v2f a;
    a.x = xr[kA];
    a.y = xr[kA + 1];
#pragma unroll
    for (int nb = 0; nb < 4; ++nb) {
      int col = nb * 16 + l;
      v2f b;
      b.x = W1[col * 128 + kA];
      b.y = W1[col * 128 + kA + 1];
      acc[nb] = __builtin_amdgcn_wmma_f32_16x16x4_f32(false, a, false, b, (short)0,
                                                      acc[nb], false, false);
    }
  }
  if (m0 + 16 <= N) {           // uniform fast path: full tile, no per-store guards
#pragma unroll
    for (int nb = 0; nb < 4; ++nb) {
#pragma unroll
      for (int r = 0; r < 8; ++r) {
        int m = m0 + r + 8 * half;
        z1[(size_t)m * 64 + nb * 16 + l] = acc[nb][r] * s1[m];
      }
    }
  } else {
#pragma unroll
    for (int nb = 0; nb < 4; ++nb) {
#pragma unroll
      for (int r = 0; r < 8; ++r) {
        int m = m0 + r + 8 * half;
        if (m < N) z1[(size_t)m * 64 + nb * 16 + l] = acc[nb][r] * s1[m];
      }
    }
  }
}

// ------------- post1: mv=proj(expmap0(z)); mobius_add(mv,bias); then logmap0 -> xt -------------
__global__ void post64_kernel(const float* __restrict__ z, const float* __restrict__ ubg,
                              float* __restrict__ xt, int N) {
  int node = blockIdx.x * (blockDim.x >> 5) + (threadIdx.x >> 5);
  if (node >= N) return;
  int lane = threadIdx.x & 31;
  bool sp = (lane != 0);                       // slot-a is spatial iff lane!=0
  const float* u = z + (size_t)node * 64;
  float ua = u[lane], ubv = u[lane + 32];
  float ba = ubg[lane], bb = ubg[lane + 32];
  // proj(expmap0(u))
  float ss  = wsum((sp ? ua * ua : 0.0f) + ubv * ubv);
  float xn  = fmaxf(sqrtf(ss), MIN_NORM);
  float sc  = sinhf(xn) / xn;
  float mva = sp ? sc * ua : 0.0f;
  float mvb = sc * ubv;
  float ss2 = wsum(mva * mva + mvb * mvb);
  float mv0 = sqrtf(fmaxf(1.0f + ss2, EPSF));
  // ptransp0(mv, ub)
  float yn2 = fmaxf(sqrtf(ss2), MIN_NORM);
  float yha = mva / yn2, yhb = mvb / yn2;
  float alpha = wsum(yha * ba + yhb * bb);
  float om = 1.0f - mv0;
  float wa = sp ? (ba - alpha * om * yha) : (ba + alpha * yn2);  // d==0: ub0 - alpha*(-yn)
  float wb = bb - alpha * om * yhb;
  // proj_tan(w, mv)
  float ux = wsum((sp ? mva * wa : 0.0f) + mvb * wb);
  float v0 = ux / fmaxf(mv0, MIN_NORM);
  float va = sp ? wa : v0;
  float vb = wb;
  // expmap(v, mv)
  float vsp   = wsum((sp ? va * va : 0.0f) + vb * vb);
  float mink  = vsp - v0 * v0;
  float normu = fminf(sqrtf(fmaxf(mink, MIN_NORM)), 1000000.0f);
  float th    = fmaxf(normu, MIN_NORM);
  float ch = coshf(th), sh = sinhf(th) / th;
  float mfa = sp ? mva : mv0;
  float ra = ch * mfa + sh * va;
  float rb = ch * mvb + sh * vb;
  // proj + (outer proj is idempotent) + logmap0
  float rss = wsum((sp ? ra * ra : 0.0f) + rb * rb);
  float h0  = sqrtf(fmaxf(1.0f + rss, EPSF));
  float ynh = fmaxf(sqrtf(rss), MIN_NORM);
  float tht = fmaxf(h0, 1.0f + EPSF);
  float ac  = logf(tht + sqrtf(fmaxf(tht * tht - 1.0f, MIN_NORM)));
  xt[(size_t)node * 64 + lane]      = sp ? ac * ra / ynh : 0.0f;
  xt[(size_t)node * 64 + lane + 32] = ac * rb / ynh;
}

// ------------- degree (shared between both aggregations) -------------
__global__ void deg_kernel(const long long* __restrict__ ei, float* __restrict__ deg, int E) {
  int e = blockIdx.x * blockDim.x + threadIdx.x;
  if (e >= E) return;
  long long t = ei[(size_t)E + e];
  atomicAdd(&deg[t], 1.0f);
}

// ------------- scatter-add, 64 dims -------------
__global__ void scatter64_kernel(const long long* __restrict__ ei, const float* __restrict__ xt,
                                 float* __restrict__ agg, int E) {
  long long idx = (long long)blockIdx.x * blockDim.x + threadIdx.x;
  int e = (int)(idx >> 6);
  int d = (int)(idx & 63);
  if (e >= E) return;
  long long s = ei[e];
  long long t = ei[(size_t)E + e];
  atomicAdd(&agg[(size_t)t * 64 + d], xt[(size_t)s * 64 + d]);
}

// ------------- agg normalize + proj(expmap0) + hyp_act + logmap0 -> t2 -------------
__global__ void aggact_kernel(const float* __restrict__ agg, const float* __restrict__ deg,
                              float* __restrict__ t2, int N) {
  int node = blockIdx.x * (blockDim.x >> 5) + (threadIdx.x >> 5);
  if (node >= N) return;
  int lane = threadIdx.x & 31;
  bool sp = (lane != 0);
  float dg = fmaxf(deg[node], 1.0f);
  float aa = agg[(size_t)node * 64 + lane] / dg;
  float ab = agg[(size_t)node * 64 + lane + 32] / dg;
  // proj(expmap0(a))
  float ss = wsum((sp ? aa * aa : 0.0f) + ab * ab);
  float xn = fmaxf(sqrtf(ss), MIN_NORM);
  float sc = sinhf(xn) / xn;
  float ea = sp ? sc * aa : 0.0f;
  float eb = sc * ab;
  float ss2 = wsum(ea * ea + eb * eb);
  float e0  = sqrtf(fmaxf(1.0f + ss2, EPSF));
  // logmap0 -> relu
  float yn1 = fmaxf(sqrtf(ss2), MIN_NORM);
  float th1 = fmaxf(e0, 1.0f + EPSF);
  float ac1 = logf(th1 + sqrtf(fmaxf(th1 * th1 - 1.0f, MIN_NORM)));
  float la = sp ? ac1 * ea / yn1 : 0.0f;
  float lb = ac1 * eb / yn1;
  float raa = fmaxf(la, 0.0f), rab = fmaxf(lb, 0.0f);
  // proj(expmap0(r))
  float ss3 = wsum((sp ? raa * raa : 0.0f) + rab * rab);
  float xn3 = fmaxf(sqrtf(ss3), MIN_NORM);
  float sc3 = sinhf(xn3) / xn3;
  float e2a = sp ? sc3 * raa : 0.0f;
  float e2b = sc3 * rab;
  float ss4 = wsum(e2a * e2a + e2b * e2b);
  float e20 = sqrtf(fmaxf(1.0f + ss4, EPSF));
  // logmap0 -> t2
  float yn4 = fmaxf(sqrtf(ss4), MIN_NORM);
  float th4 = fmaxf(e20, 1.0f + EPSF);
  float ac4 = logf(th4 + sqrtf(fmaxf(th4 * th4 - 1.0f, MIN_NORM)));
  t2[(size_t)node * 64 + lane]      = sp ? ac4 * e2a / yn4 : 0.0f;
  t2[(size_t)node * 64 + lane + 32] = ac4 * e2b / yn4;
}

// ------------- GEMM2: z2 = t2 @ W2^T (64 -> 8) via fp32 WMMA, N-padded to 16 -------------
__global__ void gemm2_kernel(const float* __restrict__ t2, const float* __restrict__ W2,
                             float* __restrict__ z2, int N) {
  int m0   = blockIdx.x * 16;
  int lane = threadIdx.x;
  int half = lane >> 4;
  int l    = lane & 15;
  int row  = m0 + l;
  int rowc = row < N ? row : N - 1;
  const float* tr = t2 + (size_t)rowc * 64;
  int   colc  = (l < 8) ? l : 0;
  float cmask = (l < 8) ? 1.0f : 0.0f;
  v8f acc = {};
#pragma unroll 2
  for (int k = 0; k < 64; k += 4) {
    int kA = k + 2 * half;
    v2f a;
    a.x = tr[kA];
    a.y = tr[kA + 1];
    v2f b;
    b.x = W2[colc * 64 + kA] * cmask;
    b.y = W2[colc * 64 + kA + 1] * cmask;
    acc = __builtin_amdgcn_wmma_f32_16x16x4_f32(false, a, false, b, (short)0,
                                                acc, false, false);
  }
  if (l < 8) {
    if (m0 + 16 <= N) {
#pragma unroll
      for (int r = 0; r < 8; ++r) {
        int m = m0 + r + 8 * half;
        z2[(size_t)m * 8 + l] = acc[r];
      }
    } else {
#pragma unroll
      for (int r = 0; r < 8; ++r) {
        int m = m0 + r + 8 * half;
        if (m < N) z2[(size_t)m * 8 + l] = acc[r];
      }
    }
  }
}

// ------------- post2: hyp_linear tail (8-dim) + logmap0 -> xt2, thread per node -------------
__global__ void post8_kernel(const float* __restrict__ z, const float* __restrict__ ubg,
                             float* __restrict__ xt, int N) {
  int n = blockIdx.x * blockDim.x + threadIdx.x;
  if (n >= N) return;
  float u[8], ub[8];
#pragma unroll
  for (int d = 0; d < 8; ++d) { u[d] = z[(size_t)n * 8 + d]; ub[d] = ubg[d]; }
  float ss = 0.f;
#pragma unroll
  for (int d = 1; d < 8; ++d) ss += u[d] * u[d];
  float xn = fmaxf(sqrtf(ss), MIN_NORM);
  float sc = sinhf(xn) / xn;
  float mv[8]; mv[0] = 0.f;
  float ss2 = 0.f;
#pragma unroll
  for (int d = 1; d < 8; ++d) { mv[d] = sc * u[d]; ss2 += mv[d] * mv[d]; }
  float mv0 = sqrtf(fmaxf(1.0f + ss2, EPSF));
  float yn2 = fmaxf(sqrtf(ss2), MIN_NORM);
  float alpha = 0.f;
#pragma unroll
  for (int d = 1; d < 8; ++d) alpha += (mv[d] / yn2) * ub[d];
  float om = 1.0f - mv0;
  float w[8];
  w[0] = ub[0] + alpha * yn2;
#pragma unroll
  for (int d = 1; d < 8; ++d) w[d] = ub[d] - alpha * om * (mv[d] / yn2);
  float ux = 0.f;
#pragma unroll
  for (int d = 1; d < 8; ++d) ux += mv[d] * w[d];
  float v0 = ux / fmaxf(mv0, MIN_NORM);
  float vsp = 0.f;
#pragma unroll
  for (int d = 1; d < 8; ++d) vsp += w[d] * w[d];
  float mink  = vsp - v0 * v0;
  float normu = fminf(sqrtf(fmaxf(mink, MIN_NORM)), 1000000.0f);
  float th = fmaxf(normu, MIN_NORM);
  float ch = coshf(th), sh = sinhf(th) / th;
  float res[8];
  res[0] = ch * mv0 + sh * v0;
  float rss = 0.f;
#pragma unroll
  for (int d = 1; d < 8; ++d) { res[d] = ch * mv[d] + sh * w[d]; rss += res[d] * res[d]; }
  float h0  = sqrtf(fmaxf(1.0f + rss, EPSF));
  float ynh = fmaxf(sqrtf(rss), MIN_NORM);
  float tht = fmaxf(h0, 1.0f + EPSF);
  float ac  = logf(tht + sqrtf(fmaxf(tht * tht - 1.0f, MIN_NORM)));
  xt[(size_t)n * 8 + 0] = 0.f;
#pragma unroll
  for (int d = 1; d < 8; ++d) xt[(size_t)n * 8 + d] = ac * res[d] / ynh;
}

// ------------- scatter-add, 8 dims -------------
__global__ void scatter8_kernel(const long long* __restrict__ ei, const float* __restrict__ xt,
                                float* __restrict__ agg, int E) {
  long long idx = (long long)blockIdx.x * blockDim.x + threadIdx.x;
  int e = (int)(idx >> 3);
  int d = (int)(idx & 7);
  if (e >= E) return;
  long long s = ei[e];
  long long t = ei[(size_t)E + e];
  atomicAdd(&agg[(size_t)t * 8 + d], xt[(size_t)s * 8 + d]);
}

// ------------- final per-node: agg2 -> hyp_agg tail -> logmap0 -> pooled/cnt -------------
__global__ void final_kernel(const float* __restrict__ agg2, const float* __restrict__ deg,
                             const long long* __restrict__ batch, float* __restrict__ pooled,
                             float* __restrict__ cnt, int N) {
  int n = blockIdx.x * blockDim.x + threadIdx.x;
  if (n >= N) return;
  float dg = fmaxf(deg[n], 1.0f);
  float a[8];
#pragma unroll
  for (int d = 0; d < 8; ++d) a[d] = agg2[(size_t)n * 8 + d] / dg;
  float ss = 0.f;
#pragma unroll
  for (int d = 1; d < 8; ++d) ss += a[d] * a[d];
  float xn = fmaxf(sqrtf(ss), MIN_NORM);
  float sc = sinhf(xn) / xn;
  float e[8]; float ss2 = 0.f;
#pragma unroll
  for (int d = 1; d < 8; ++d) { e[d] = sc * a[d]; ss2 += e[d] * e[d]; }
  float e0  = sqrtf(fmaxf(1.0f + ss2, EPSF));
  float yn  = fmaxf(sqrtf(ss2), MIN_NORM);
  float tht = fmaxf(e0, 1.0f + EPSF);
  float ac  = logf(tht + sqrtf(fmaxf(tht * tht - 1.0f, MIN_NORM)));
  int g = (int)batch[n];
  atomicAdd(&cnt[g], 1.0f);
#pragma unroll
  for (int d = 1; d < 8; ++d) atomicAdd(&pooled[g * 8 + d], ac * e[d] / yn);
  // component 0 of the tangent is exactly 0 -> pooled[g*8+0] stays zeroed
}

// ------------- log_softmax over 64 graphs x 8 classes -------------
__global__ void softmax_kernel(const float* __restrict__ pooled, const float* __restrict__ cnt,
                               float* __restrict__ out, int G) {
  int g = blockIdx.x * blockDim.x + threadIdx.x;
  if (g >= G) return;
  float c = fmaxf(cnt[g], 1.0f);
  float p[8]; float m = -1e30f;
#pragma unroll
  for (int d = 0; d < 8; ++d) { p[d] = pooled[g * 8 + d] / c; m = fmaxf(m, p[d]); }
  float s = 0.f;
#pragma unroll
  for (int d = 0; d < 8; ++d) s += expf(p[d] - m);
  float lse = m + logf(s);
#pragma unroll
  for (int d = 0; d < 8; ++d) out[g * 8 + d] = p[d] - lse;
}

extern "C" void kernel_launch(void* const* d_in, const int* in_sizes, int n_in,
                              void* d_out, int out_size, void* d_ws, size_t ws_size,
                              hipStream_t stream) {
  const float*     x     = (const float*)d_in[0];
  const float*     W1    = (const float*)d_in[1];
  const float*     b1    = (const float*)d_in[2];
  const float*     W2    = (const float*)d_in[3];
  const float*     b2    = (const float*)d_in[4];
  const long long* ei    = (const long long*)d_in[5];  // int64, shape (2,E) flat
  const long long* batch = (const long long*)d_in[6];  // int64, shape (N,)
  float* out = (float*)d_out;

  const int N = in_sizes[6];
  const int E = in_sizes[5] / 2;
  const int G = out_size / 8;

  float* ws = (float*)d_ws;
  size_t o = 0;
  float* ub1    = ws + o; o += 64;
  float* ub2    = ws + o; o += 64;
  float* pooled = ws + o; o += 512;              // accumulator region starts here
  float* cnt    = ws + o; o += 64;
  float* deg    = ws + o; o += (size_t)N;
  float* agg1   = ws + o; o += (size_t)N * 64;
  float* agg2   = ws + o; o += (size_t)N * 8;
  const size_t zcount = 512 + 64 + (size_t)N * 73;  // pooled..agg2 contiguous
  float* s1  = ws + o; o += (size_t)N;
  float* z1  = ws + o; o += (size_t)N * 64;         // reused as t2
  float* xt1 = ws + o; o += (size_t)N * 64;         // reused as z2/xt2
  float* t2  = z1;
  float* z2  = xt1;
  float* xt2 = xt1 + (size_t)N * 8;

  // 1. zero all accumulators (graph-replay safe)
  zero_kernel<<<(unsigned)((zcount + 255) / 256), 256, 0, stream>>>(pooled, zcount);
  // 2. layer-constant bias tangents
  bias_kernel<<<1, 1, 0, stream>>>(b1, ub1, 64);
  bias_kernel<<<1, 1, 0, stream>>>(b2, ub2, 8);
  // 3. per-node logmap0 scale
  row_scale_kernel<<<(N + 7) / 8, 256, 0, stream>>>(x, s1, N);
  // 4. GEMM1 (fp32 WMMA) with logmap0 folded
  gemm1_kernel<<<(N + 15) / 16, 32, 0, stream>>>(x, W1, s1, z1, N);
  // 5. hyp_linear tail + hyp_agg's logmap0
  post64_kernel<<<(N + 7) / 8, 256, 0, stream>>>(z1, ub1, xt1, N);
  // 6. degree counts (shared by both aggregations)
  deg_kernel<<<(E + 255) / 256, 256, 0, stream>>>(ei, deg, E);
  // 7. edge scatter-add, 64 dims
  {
    long long tot = (long long)E * 64;
    scatter64_kernel<<<(unsigned)((tot + 255) / 256), 256, 0, stream>>>(ei, xt1, agg1, E);
  }
  // 8. agg normalize + expmap0 + hyp_act + logmap0
  aggact_kernel<<<(N + 7) / 8, 256, 0, stream>>>(agg1, deg, t2, N);
  // 9. GEMM2 (fp32 WMMA), 64 -> 8
  gemm2_kernel<<<(N + 15) / 16, 32, 0, stream>>>(t2, W2, z2, N);
  // 10. hyp_linear tail (8-dim) + logmap0
  post8_kernel<<<(N + 255) / 256, 256, 0, stream>>>(z2, ub2, xt2, N);
  // 11. edge scatter-add, 8 dims
  {
    long long tot = (long long)E * 8;
    scatter8_kernel<<<(unsigned)((tot + 255) / 256), 256, 0, stream>>>(ei, xt2, agg2, E);
  }
  // 12. final hyp_agg tail + graph pooling
  final_kernel<<<(N + 255) / 256, 256, 0, stream>>>(agg2, deg, batch, pooled, cnt, N);
  // 13. log_softmax
  softmax_kernel<<<(G + 63) / 64, 64, 0, stream>>>(pooled, cnt, out, G);
}